// ConnectionsSMP_83760452206682
// MI455X (gfx1250) — compile-verified
//
#include <hip/hip_runtime.h>
#include <math.h>

#define BATCH 8
#define FDIM  64
#define HWPIX (512*512)
#define NINST 32
#define TILE  4096
#define K1T   256

typedef float v2f __attribute__((ext_vector_type(2)));
typedef float v8f __attribute__((ext_vector_type(8)));

// ---------------- Kernel 0: zero the vectors region of d_out -----------------
__global__ void zero_vectors_kernel(float* __restrict__ out) {
    int i = blockIdx.x * blockDim.x + threadIdx.x;
    if (i < BATCH * NINST * FDIM) out[i] = 0.0f;
}

// ---------------- Kernel 1: streaming segment-max (bandwidth bound) ----------
// vectors[b][n][f] = max(0, max over pixels with id==n+1 of encoded[b][f][pix])
// Int-punned atomic max is exact because stored values are always >= +0.0f.
__global__ void segmax_kernel(const float* __restrict__ encoded,
                              const int*   __restrict__ masks,
                              float*       __restrict__ vectors) {
    __shared__ int sid[TILE];          // instance ids for this pixel tile
    __shared__ int smax[FDIM * NINST]; // f-major partial maxima (int-punned floats)

    const int tid = threadIdx.x;
    const int tilesPerBatch = HWPIX / TILE;
    const int b    = blockIdx.x / tilesPerBatch;
    const int tile = blockIdx.x % tilesPerBatch;

    for (int e = tid; e < FDIM * NINST; e += K1T) smax[e] = 0;

    // load the tile's mask ids once (they are reused for all 64 features)
    const int4* m4 = (const int4*)(masks + (long)b * HWPIX + (long)tile * TILE);
    int4* s4 = (int4*)sid;
    #pragma unroll
    for (int u = 0; u < TILE / 4 / K1T; ++u)
        s4[u * K1T + tid] = m4[u * K1T + tid];
    __syncthreads();

    for (int f = 0; f < FDIM; ++f) {
        const float4* e4 = (const float4*)(encoded +
            ((long)(b * FDIM + f)) * HWPIX + (long)tile * TILE);
        int* srow = smax + f * NINST;   // bank pattern (n + 32*f) % 64 : conflict-free per f
        #pragma unroll
        for (int u = 0; u < TILE / 4 / K1T; ++u) {
            int idx  = u * K1T + tid;
            float4 v = e4[idx];
            int4   id = s4[idx];
            if (id.x > 0) atomicMax(&srow[id.x - 1], __float_as_int(v.x));
            if (id.y > 0) atomicMax(&srow[id.y - 1], __float_as_int(v.y));
            if (id.z > 0) atomicMax(&srow[id.z - 1], __float_as_int(v.z));
            if (id.w > 0) atomicMax(&srow[id.w - 1], __float_as_int(v.w));
        }
    }
    __syncthreads();

    // flush non-trivial partials to global (vectors pre-zeroed)
    for (int e = tid; e < FDIM * NINST; e += K1T) {
        int f = e / NINST, n = e % NINST;
        int v = smax[e];
        if (v > 0)
            atomicMax((int*)&vectors[(b * NINST + n) * FDIM + f], v);
    }
}

// ---------------- Kernel 2: collapse the two linear layers -------------------
// Wc[64][16] : cols 0..3 = w1[0:64]@w2 (U), cols 4..7 = w1[64:128]@w2 (T), rest 0
// r[4]       : b1@w2 + b2
__global__ void precompute_kernel(const float* __restrict__ w1,
                                  const float* __restrict__ b1,
                                  const float* __restrict__ w2,
                                  const float* __restrict__ b2,
                                  float* __restrict__ ws) {
    int k = threadIdx.x;               // 0..63
    float acc[8];
    #pragma unroll
    for (int c = 0; c < 8; ++c) acc[c] = 0.0f;
    for (int h = 0; h < 32; ++h) {
        float a  = w1[k * 32 + h];          // top half row k
        float bb = w1[(64 + k) * 32 + h];   // bottom half row k
        #pragma unroll
        for (int c = 0; c < 4; ++c) {
            float w = w2[h * 4 + c];
            acc[c]     += a  * w;
            acc[4 + c] += bb * w;
        }
    }
    #pragma unroll
    for (int c = 0; c < 8; ++c)  ws[k * 16 + c] = acc[c];
    #pragma unroll
    for (int c = 8; c < 16; ++c) ws[k * 16 + c] = 0.0f;   // N-padding for WMMA
    if (k == 0) {
        for (int c = 0; c < 4; ++c) {
            float r = b2[c];
            for (int h = 0; h < 32; ++h) r += b1[h] * w2[h * 4 + c];
            ws[1024 + c] = r;
        }
    }
}

// ---------------- Kernel 3: per-batch 32x64x8 GEMM on WMMA + sigmoid ---------
// PQ = V(32x64) @ Wc(64x16) via V_WMMA_F32_16X16X4_F32, then
// connections[b,c,j,i] = sigmoid(PQ[i,c] + PQ[j,4+c] + r[c])
__global__ void relation_wmma_kernel(const float* __restrict__ vectors,
                                     const float* __restrict__ ws,
                                     float* __restrict__ conn) {
    __shared__ float Vs[32 * 64];
    __shared__ float Wcs[64 * 16];
    __shared__ float PQ[32 * 16];
    __shared__ float rs[4];

    const int lane = threadIdx.x;      // 0..31, one wave per batch
    const int b    = blockIdx.x;
    const float* vb = vectors + b * NINST * FDIM;

    for (int i = lane; i < 32 * 64; i += 32) Vs[i]  = vb[i];
    for (int i = lane; i < 64 * 16; i += 32) Wcs[i] = ws[i];
    if (lane < 4) rs[lane] = ws[1024 + lane];
    __syncthreads();

    const int l  = lane & 15;
    const int hi = lane >> 4;          // 0: K pair {0,1} / rows M, 1: K pair {2,3} / rows M+8

    v8f acc0 = {0,0,0,0,0,0,0,0};      // rows 0..15
    v8f acc1 = {0,0,0,0,0,0,0,0};      // rows 16..31
    #pragma unroll
    for (int k = 0; k < 16; ++k) {     // K = 64 in steps of 4
        const int k0 = 4 * k + 2 * hi;
        v2f bf; bf.x = Wcs[k0 * 16 + l];        bf.y = Wcs[(k0 + 1) * 16 + l];
        v2f a0; a0.x = Vs[l * 64 + k0];         a0.y = Vs[l * 64 + k0 + 1];
        v2f a1; a1.x = Vs[(16 + l) * 64 + k0];  a1.y = Vs[(16 + l) * 64 + k0 + 1];
        acc0 = __builtin_amdgcn_wmma_f32_16x16x4_f32(
                   false, a0, false, bf, (short)0, acc0, false, false);
        acc1 = __builtin_amdgcn_wmma_f32_16x16x4_f32(
                   false, a1, false, bf, (short)0, acc1, false, false);
    }

    #pragma unroll
    for (int r = 0; r < 8; ++r) {      // C/D layout: VGPR r, lane>=16 -> row r+8
        PQ[(r + 8 * hi) * 16 + l]        = acc0[r];
        PQ[(16 + r + 8 * hi) * 16 + l]   = acc1[r];
    }
    __syncthreads();

    float Pi[4];
    #pragma unroll
    for (int c = 0; c < 4; ++c) Pi[c] = PQ[lane * 16 + c];

    float* ob = conn + b * 4 * NINST * NINST;   // layout [c][j][i], i = lane (coalesced)
    for (int c = 0; c < 4; ++c) {
        float base = rs[c] + Pi[c];
        for (int j = 0; j < 32; ++j) {
            float x = base + PQ[j * 16 + 4 + c];
            float s = 1.0f / (1.0f + __expf(-x));
            ob[(c * NINST + j) * NINST + lane] = s;
        }
    }
}

// -----------------------------------------------------------------------------
extern "C" void kernel_launch(void* const* d_in, const int* in_sizes, int n_in,
                              void* d_out, int out_size, void* d_ws, size_t ws_size,
                              hipStream_t stream) {
    (void)in_sizes; (void)n_in; (void)out_size; (void)ws_size;
    const float* encoded = (const float*)d_in[0];
    const int*   masks   = (const int*)  d_in[1];
    const float* w1      = (const float*)d_in[2];
    const float* b1      = (const float*)d_in[3];
    const float* w2      = (const float*)d_in[4];
    const float* b2      = (const float*)d_in[5];
    float* out = (float*)d_out;          // [0,16384): vectors ; [16384,49152): connections
    float* ws  = (float*)d_ws;

    zero_vectors_kernel<<<(BATCH * NINST * FDIM + 255) / 256, 256, 0, stream>>>(out);
    segmax_kernel<<<BATCH * (HWPIX / TILE), K1T, 0, stream>>>(encoded, masks, out);
    precompute_kernel<<<1, 64, 0, stream>>>(w1, b1, w2, b2, ws);
    relation_wmma_kernel<<<BATCH, 32, 0, stream>>>(out, ws, out + BATCH * NINST * FDIM);
}